// GCN_71262097376127
// MI455X (gfx1250) — compile-verified
//
#include <hip/hip_runtime.h>
#include <hip/hip_bf16.h>

typedef float v2f __attribute__((ext_vector_type(2)));
typedef float v8f __attribute__((ext_vector_type(8)));

#define NNODES 100000
#define NEDGES 1600000
#define HF 128
#define NCLS 8
#define NLAYERS 6
#define XSTR 132   // padded LDS row stride (floats) to avoid 16-way bank conflicts

// ---------------- elementwise helpers ----------------

__global__ void zero_f(float* __restrict__ p, int n) {
  int i = blockIdx.x * blockDim.x + threadIdx.x;
  if (i < n) p[i] = 0.0f;
}

__global__ void degree_kernel(const int* __restrict__ src, const int* __restrict__ dst,
                              float* __restrict__ outDeg, float* __restrict__ inDeg, int e) {
  int i = blockIdx.x * blockDim.x + threadIdx.x;
  if (i < e) {
    unsafeAtomicAdd(&outDeg[src[i]], 1.0f);
    unsafeAtomicAdd(&inDeg[dst[i]], 1.0f);
  }
}

__global__ void norm_kernel(float* __restrict__ a, float* __restrict__ b, int n) {
  int i = blockIdx.x * blockDim.x + threadIdx.x;
  if (i < n) {
    a[i] = rsqrtf(fmaxf(a[i], 1.0f));
    b[i] = rsqrtf(fmaxf(b[i], 1.0f));
  }
}

// ---------------- fp32 WMMA GEMM: Y = (X @ W) * norm_src[row] ----------------
// One wave computes a 16-row x 128-col stripe: 8 accumulator tiles of 16x16,
// K-loop of 32 x V_WMMA_F32_16X16X4_F32. W staged in LDS (64KB), the wave's
// 16 A-rows staged in LDS with padded stride.
__global__ __launch_bounds__(128) void gemm_norm_wmma(
    const float* __restrict__ X,      // [nRows,128]
    const float* __restrict__ W,      // [128,128] row-major (k,n)
    const float* __restrict__ normSrc,// [nRows]
    float* __restrict__ Y,            // [nRows,128]
    int nRows)
{
  extern __shared__ float smem[];
  float* sW = smem;                 // 128*128 floats
  float* sX = smem + HF * HF;       // 4 waves * 16 rows * XSTR floats

  // cooperative load of W into LDS (float4)
  {
    const float4* Wv = (const float4*)W;
    float4* sWv = (float4*)sW;
    for (int i = threadIdx.x; i < HF * HF / 4; i += blockDim.x) sWv[i] = Wv[i];
  }

  const int wave = threadIdx.x >> 5;
  const int lane = threadIdx.x & 31;
  const int tile = blockIdx.x * 4 + wave;   // 16-row tile index
  const int row0 = tile * 16;
  float* sXw = sX + wave * 16 * XSTR;

  if (row0 < nRows) {
    // each j: one full row, 32 lanes cover 32 float4 = 128 floats, coalesced
    for (int j = 0; j < 16; ++j) {
      float4 v = ((const float4*)(X + (size_t)(row0 + j) * HF))[lane];
      *(float4*)(sXw + j * XSTR + lane * 4) = v;
    }
  }
  __syncthreads();
  if (row0 >= nRows) return;

  const int half = lane >> 4;   // 0: K+{0,1}, 1: K+{2,3}
  const int l16  = lane & 15;
  const float* aRow = sXw + l16 * XSTR;   // M = lane%16

  v8f acc[8];
  v8f vz = {};
  #pragma unroll
  for (int c = 0; c < 8; ++c) acc[c] = vz;

  for (int k0 = 0; k0 < HF; k0 += 4) {
    v2f a;
    a.x = aRow[k0 + 2 * half];
    a.y = aRow[k0 + 2 * half + 1];
    #pragma unroll
    for (int c = 0; c < 8; ++c) {
      v2f b;
      const int n = c * 16 + l16;
      b.x = sW[(k0 + 2 * half) * HF + n];
      b.y = sW[(k0 + 2 * half + 1) * HF + n];
      acc[c] = __builtin_amdgcn_wmma_f32_16x16x4_f32(
          false, a, false, b, (short)0, acc[c], false, false);
    }
  }

  // epilogue: D layout -> row m = row0 + r + 8*half, col = c*16 + l16
  const int mBase = row0 + 8 * half;
  float nrm[8];
  #pragma unroll
  for (int r = 0; r < 8; ++r) nrm[r] = normSrc[mBase + r];
  #pragma unroll
  for (int c = 0; c < 8; ++c) {
    #pragma unroll
    for (int r = 0; r < 8; ++r) {
      Y[(size_t)(mBase + r) * HF + c * 16 + l16] = acc[c][r] * nrm[r];
    }
  }
}

// ---------------- edge scatter: agg[dst] += hS[src] * ew ----------------
// one wave32 per edge; lane handles 4 consecutive features (float4)
__global__ void edge_scatter(const float* __restrict__ hS, const int* __restrict__ src,
                             const int* __restrict__ dst, const float* __restrict__ ew,
                             float* __restrict__ agg, int nEdges) {
  int gid = blockIdx.x * blockDim.x + threadIdx.x;
  int e = gid >> 5;
  int lane = gid & 31;
  if (e >= nEdges) return;
  int s = src[e];
  int d = dst[e];
  float w = ew[e];
  float4 v = ((const float4*)(hS + (size_t)s * HF))[lane];
  float* ap = agg + (size_t)d * HF + lane * 4;
  unsafeAtomicAdd(ap + 0, v.x * w);
  unsafeAtomicAdd(ap + 1, v.y * w);
  unsafeAtomicAdd(ap + 2, v.z * w);
  unsafeAtomicAdd(ap + 3, v.w * w);
}

// ---------------- finalize: h = leaky_relu(agg * norm_dst + b) ----------------
__global__ void finalize_kernel(const float* __restrict__ agg, const float* __restrict__ normDst,
                                const float* __restrict__ bias, float* __restrict__ Hout, int n128) {
  int i = blockIdx.x * blockDim.x + threadIdx.x;
  if (i < n128) {
    int row = i >> 7;
    int f = i & 127;
    float x = agg[i] * normDst[row] + bias[f];
    Hout[i] = (x > 0.0f) ? x : 0.01f * x;
  }
}

// ---------------- final linear: out = h @ lin_W + lin_b ----------------
__global__ void final_linear(const float* __restrict__ H, const float* __restrict__ Wl,
                             const float* __restrict__ bl, float* __restrict__ out, int n) {
  int gid = blockIdx.x * blockDim.x + threadIdx.x;
  if (gid >= n * NCLS) return;
  int row = gid >> 3;
  int c = gid & 7;
  const float* h = H + (size_t)row * HF;
  float s = bl[c];
  #pragma unroll 8
  for (int f = 0; f < HF; ++f) s = fmaf(h[f], Wl[f * NCLS + c], s);
  out[gid] = s;
}

// ---------------- launcher ----------------
extern "C" void kernel_launch(void* const* d_in, const int* in_sizes, int n_in,
                              void* d_out, int out_size, void* d_ws, size_t ws_size,
                              hipStream_t stream) {
  const float* in_feat = (const float*)d_in[0];
  const int*   src     = (const int*)d_in[1];
  const int*   dst     = (const int*)d_in[2];
  const float* ew      = (const float*)d_in[3];
  const float* convW   = (const float*)d_in[4];   // [6][128][128]
  const float* convB   = (const float*)d_in[5];   // [6][128]
  const float* linW    = (const float*)d_in[6];   // [128][8]
  const float* linB    = (const float*)d_in[7];   // [8]
  float* out = (float*)d_out;

  float* ws = (float*)d_ws;
  const size_t NF = (size_t)NNODES * HF;          // 12.8M floats
  float* hA      = ws;                            // layer output / next input
  float* hS      = ws + NF;                       // scaled GEMM output
  float* agg     = ws + 2 * NF;                   // scatter accumulator
  float* normSrc = ws + 3 * NF;                   // also out-degree scratch
  float* normDst = normSrc + NNODES;              // also in-degree scratch

  const int nf = (int)NF;

  // degree norms (structural, computed once)
  zero_f<<<(2 * NNODES + 255) / 256, 256, 0, stream>>>(normSrc, 2 * NNODES);
  degree_kernel<<<(NEDGES + 255) / 256, 256, 0, stream>>>(src, dst, normSrc, normDst, NEDGES);
  norm_kernel<<<(NNODES + 255) / 256, 256, 0, stream>>>(normSrc, normDst, NNODES);

  const size_t ldsBytes = (size_t)(HF * HF + 4 * 16 * XSTR) * sizeof(float); // ~97KB
  const int gemmBlocks = (NNODES / 16 + 3) / 4;   // 4 waves (16-row tiles) per block
  const int scatterBlocks = (int)(((size_t)NEDGES * 32 + 255) / 256);

  for (int l = 0; l < NLAYERS; ++l) {
    const float* x = (l == 0) ? in_feat : hA;
    gemm_norm_wmma<<<gemmBlocks, 128, ldsBytes, stream>>>(
        x, convW + (size_t)l * HF * HF, normSrc, hS, NNODES);
    zero_f<<<(nf + 255) / 256, 256, 0, stream>>>(agg, nf);
    edge_scatter<<<scatterBlocks, 256, 0, stream>>>(hS, src, dst, ew, agg, NEDGES);
    finalize_kernel<<<(nf + 255) / 256, 256, 0, stream>>>(agg, normDst, convB + l * HF, hA, nf);
  }

  final_linear<<<(NNODES * NCLS + 255) / 256, 256, 0, stream>>>(hA, linW, linB, out, NNODES);
}